// GetKnnGraph_57251914056096
// MI455X (gfx1250) — compile-verified
//
#include <hip/hip_runtime.h>
#include <hip/hip_bf16.h>

// Fixed problem shape from the reference setup_inputs(): B=4, N=8192, C=3, k=16.
#define NPTS 8192
#define KSEL 16
#define ROWS 16          // rows per workgroup tile
#define NWAVES 8         // 256 threads / wave32
#define CHUNK_COLS (NWAVES * 16)   // 128 columns staged per chunk

typedef float v2f __attribute__((ext_vector_type(2)));
typedef float v8f __attribute__((ext_vector_type(8)));

// -------- pass 1: per-point squared norm, rounding order matches jnp.sum(p*p, -1) --------
__global__ void knn_sq_kernel(const float* __restrict__ pts, float* __restrict__ sq, int total) {
    int i = blockIdx.x * blockDim.x + threadIdx.x;
    if (i < total) {
        float p0 = pts[i * 3 + 0];
        float p1 = pts[i * 3 + 1];
        float p2 = pts[i * 3 + 2];
        sq[i] = __fadd_rn(__fadd_rn(__fmul_rn(p0, p0), __fmul_rn(p1, p1)), __fmul_rn(p2, p2));
    }
}

__device__ __forceinline__ bool better(float d, int i, float d2, int i2) {
    // strict total order: smaller distance first, smaller index on ties (lax.top_k stability)
    return (d < d2) || (d == d2 && i < i2);
}

// -------- pass 2: WMMA distance tiles + top-k selection --------
__global__ __launch_bounds__(256) void knn_main_kernel(const float* __restrict__ pts,
                                                       const float* __restrict__ sq,
                                                       int* __restrict__ out,
                                                       int nBatch) {
    __shared__ float sD[ROWS * CHUNK_COLS];      // 8 KB distance staging tile
    __shared__ float candD[ROWS * 256];          // 16 KB merge candidates (dist)
    __shared__ int   candI[ROWS * 256];          // 16 KB merge candidates (idx)

    const int tid  = threadIdx.x;
    const int wave = tid >> 5;
    const int lane = tid & 31;

    const int tilesPerBatch = NPTS / ROWS;       // 512
    const int b       = blockIdx.x / tilesPerBatch;
    const int rowbase = (blockIdx.x % tilesPerBatch) * ROWS;
    const int baseIdx = b * NPTS;

    // ---- A operand: 16x4 f32 (C=3 padded). lanes 0-15: K=0,1 ; lanes 16-31: K=2,0 ----
    const int arow = rowbase + (lane & 15);
    v2f a;
    if (lane < 16) {
        a.x = pts[(size_t)(baseIdx + arow) * 3 + 0];
        a.y = pts[(size_t)(baseIdx + arow) * 3 + 1];
    } else {
        a.x = pts[(size_t)(baseIdx + arow) * 3 + 2];
        a.y = 0.0f;
    }

    // sa[j] = ||row||^2 for D-matrix VGPR j (M = j for lanes 0-15, j+8 for lanes 16-31)
    const int mofs = (lane >> 4) * 8;
    float sa[8];
#pragma unroll
    for (int j = 0; j < 8; ++j)
        sa[j] = sq[baseIdx + rowbase + j + mofs];

    // ---- per-thread selection state: sorted 16-entry list over a column partition ----
    const int selRow = tid >> 4;   // 0..15: which row of the tile this thread selects for
    const int selSub = tid & 15;   // 16 selector threads per row
    float bd[KSEL];
    int   bi[KSEL];
#pragma unroll
    for (int i = 0; i < KSEL; ++i) { bd[i] = 3.402823466e38f; bi[i] = 0x7fffffff; }

    for (int chunk = 0; chunk < NPTS / CHUNK_COLS; ++chunk) {   // 64 chunks
        // ---- compute phase: each wave produces one 16x16 tile via WMMA ----
        const int colTileBase = chunk * CHUNK_COLS + wave * 16;
        const int col = colTileBase + (lane & 15);
        v2f bm;
        if (lane < 16) {
            bm.x = pts[(size_t)(baseIdx + col) * 3 + 0];
            bm.y = pts[(size_t)(baseIdx + col) * 3 + 1];
        } else {
            bm.x = pts[(size_t)(baseIdx + col) * 3 + 2];
            bm.y = 0.0f;
        }
        const float sb = sq[baseIdx + colTileBase + (lane & 15)];

        v8f c = {};
        // D = A(16x4) * B(4x16) + 0   -> dot(x_m, y_n) per element
        c = __builtin_amdgcn_wmma_f32_16x16x4_f32(
                /*neg_a=*/false, a, /*neg_b=*/false, bm,
                /*c_mod=*/(short)0, c, /*reuse_a=*/false, /*reuse_b=*/false);

        const int lcol = wave * 16 + (lane & 15);
#pragma unroll
        for (int j = 0; j < 8; ++j) {
            // d = (||x||^2 + ||y||^2) - 2*dot, same rounding structure as the reference
            float d = __fsub_rn(__fadd_rn(sa[j], sb), __fmul_rn(2.0f, c[j]));
            sD[(j + mofs) * CHUNK_COLS + lcol] = d;
        }
        __syncthreads();

        // ---- selection phase: each thread scans 8 columns of its row ----
        const int gcolbase = chunk * CHUNK_COLS + selSub * 8;
#pragma unroll
        for (int j = 0; j < 8; ++j) {
            float d  = sD[selRow * CHUNK_COLS + selSub * 8 + j];
            int  idx = gcolbase + j;
            if (better(d, idx, bd[KSEL - 1], bi[KSEL - 1])) {
                bd[KSEL - 1] = d; bi[KSEL - 1] = idx;
#pragma unroll
                for (int q = KSEL - 2; q >= 0; --q) {
                    if (better(bd[q + 1], bi[q + 1], bd[q], bi[q])) {
                        float td = bd[q]; bd[q] = bd[q + 1]; bd[q + 1] = td;
                        int   ti = bi[q]; bi[q] = bi[q + 1]; bi[q + 1] = ti;
                    }
                }
            }
        }
        __syncthreads();
    }

    // ---- final merge: 16 sorted lists (16 entries each) per row -> top-16 ----
#pragma unroll
    for (int i = 0; i < KSEL; ++i) {
        candD[selRow * 256 + selSub * KSEL + i] = bd[i];
        candI[selRow * 256 + selSub * KSEL + i] = bi[i];
    }
    __syncthreads();

    if (selSub == 0) {
        float fd[KSEL];
        int   fi[KSEL];
#pragma unroll
        for (int i = 0; i < KSEL; ++i) { fd[i] = 3.402823466e38f; fi[i] = 0x7fffffff; }
        for (int c2 = 0; c2 < 256; ++c2) {
            float d  = candD[selRow * 256 + c2];
            int  idx = candI[selRow * 256 + c2];
            if (better(d, idx, fd[KSEL - 1], fi[KSEL - 1])) {
                fd[KSEL - 1] = d; fi[KSEL - 1] = idx;
#pragma unroll
                for (int q = KSEL - 2; q >= 0; --q) {
                    if (better(fd[q + 1], fi[q + 1], fd[q], fi[q])) {
                        float td = fd[q]; fd[q] = fd[q + 1]; fd[q + 1] = td;
                        int   ti = fi[q]; fi[q] = fi[q + 1]; fi[q + 1] = ti;
                    }
                }
            }
        }
        // edge_index layout: out[0 .. E) = src flat, out[E .. 2E) = tgt flat, E = B*N*k
        const int n    = rowbase + selRow;
        const int srcv = baseIdx + n;
        const size_t E    = (size_t)nBatch * NPTS * KSEL;
        const size_t eoff = (size_t)srcv * KSEL;
#pragma unroll
        for (int i = 0; i < KSEL; ++i) {
            out[eoff + i]     = srcv;
            out[E + eoff + i] = baseIdx + fi[i];
        }
    }
}

extern "C" void kernel_launch(void* const* d_in, const int* in_sizes, int n_in,
                              void* d_out, int out_size, void* d_ws, size_t ws_size,
                              hipStream_t stream) {
    (void)n_in; (void)out_size; (void)ws_size;
    const float* pts = (const float*)d_in[0];
    const int totalPts = in_sizes[0] / 3;     // B * N
    const int B = totalPts / NPTS;            // N fixed at 8192 by the reference setup

    float* sqbuf = (float*)d_ws;              // B*N floats = 128 KB scratch

    knn_sq_kernel<<<(totalPts + 255) / 256, 256, 0, stream>>>(pts, sqbuf, totalPts);

    const int nBlocks = B * (NPTS / ROWS);    // 2048
    knn_main_kernel<<<nBlocks, 256, 0, stream>>>(pts, sqbuf, (int*)d_out, B);
}